// SMRNN_59064390255145
// MI455X (gfx1250) — compile-verified
//
#include <hip/hip_runtime.h>
#include <math.h>

// ---------------------------------------------------------------------------
// MI455X (gfx1250) — bandwidth-bound self-modifying-MLP evaluation.
// Dominant traffic: w2_4 (1,452,480 x 64 fp32 = 372 MB) -> ~16us @ 23.3 TB/s.
// All layers are matvecs; we use V_WMMA_F32_16X16X4_F32 (native fp32 WMMA):
// one wave per 16-row output tile, B = x broadcast into all 16 columns.
// Template specialization keeps the streaming inner loop branch-free, and the
// 372MB stream uses non-temporal loads so the ~22MB of small-layer weights
// stay resident in the 192MB L2 across graph replays.
// ---------------------------------------------------------------------------

typedef __attribute__((ext_vector_type(2))) float v2f;
typedef __attribute__((ext_vector_type(8))) float v8f;

#define LRELU_ALPHA 0.01f

// act: 0 = none, 1 = leaky_relu, 2 = 2*tanh, 3 = 5*tanh
template <bool HAS_DW, bool NT>
__global__ __launch_bounds__(256) void matvec_wmma_f32(
    const float* __restrict__ W,   // [M,K] row-major
    const float* __restrict__ dW,  // delta weights, same layout (used iff HAS_DW)
    const float* __restrict__ b,   // [M]
    const float* __restrict__ pb,  // optional extra bias [M] (or null)
    float bscale,                  // bias scale (net3 uses 2*b3 + pbc)
    const float* __restrict__ x,   // [K]
    float* __restrict__ y,         // [M]
    int M, int K, int act)
{
    __shared__ float sx[1024];  // K <= 1024 for every layer
    for (int i = threadIdx.x; i < K; i += blockDim.x) sx[i] = x[i];
    __syncthreads();

    const int lane = threadIdx.x & 31;
    const int mrow = lane & 15;                 // A-matrix row owned by this lane
    const int koff = (lane < 16) ? 0 : 2;       // ISA A layout: lanes16-31 hold K+2,K+3
    const int wave = blockIdx.x * (blockDim.x >> 5) + (threadIdx.x >> 5);
    const int nwav = gridDim.x * (blockDim.x >> 5);
    const int ntiles = M >> 4;                  // all M are multiples of 16

    for (int tile = wave; tile < ntiles; tile += nwav) {
        const size_t rowbase = (size_t)(tile * 16 + mrow) * (size_t)K + (size_t)koff;
        const float* wrow = W + rowbase;
        const float* drow = dW + rowbase;       // only dereferenced if HAS_DW

        v8f c = {};
        #pragma unroll 8
        for (int k = 0; k < K; k += 4) {
            // A tile 16x4: lane m<16 -> W[m][k],W[m][k+1]; lane m>=16 -> W[m][k+2],W[m][k+3]
            v2f a = NT ? __builtin_nontemporal_load((const v2f*)(wrow + k))
                       : *(const v2f*)(wrow + k);
            if constexpr (HAS_DW) {
                v2f d = *(const v2f*)(drow + k);
                a.x += d.x; a.y += d.y;
            }
            // B tile 4x16: every column j holds x[k..k+3] (mirrored lane layout)
            v2f bv = *(const v2f*)(&sx[k + koff]);
            c = __builtin_amdgcn_wmma_f32_16x16x4_f32(
                    /*neg_a=*/false, a, /*neg_b=*/false, bv,
                    /*c_mod=*/(short)0, c, /*reuse_a=*/false, /*reuse_b=*/false);
        }

        // D layout: lane 0 holds rows 0..7 in c[0..7]; lane 16 holds rows 8..15.
        if (lane == 0 || lane == 16) {
            const int rbase = tile * 16 + (lane >> 1);  // lane16 -> +8
            #pragma unroll
            for (int j = 0; j < 8; ++j) {
                const int r = rbase + j;
                float v = c[j] + bscale * b[r] + (pb ? pb[r] : 0.0f);
                if (act == 1)      v = (v > 0.0f) ? v : LRELU_ALPHA * v;
                else if (act == 2) v = 2.0f * tanhf(v);
                else if (act == 3) v = 5.0f * tanhf(v);
                y[r] = v;
            }
        }
    }
}

__global__ void concat_kernel(const float* __restrict__ a,
                              const float* __restrict__ p,
                              const float* __restrict__ s,
                              float* __restrict__ X)
{
    const int i = threadIdx.x;  // 256 threads
    X[i] = (i < 64) ? a[i] : (i < 128) ? p[i - 64] : s[i - 128];
}

static inline int grid_for(int M) {
    const int ntiles = M / 16;
    return (ntiles + 7) / 8;  // 8 waves/block, 1 tile/wave/iter
}

// plain layer (net1, net2 small layers)
static inline void mv(const float* W, const float* b, const float* x, float* y,
                      int M, int K, int act, hipStream_t stream) {
    matvec_wmma_f32<false, false><<<grid_for(M), 256, 0, stream>>>(
        W, nullptr, b, nullptr, 1.0f, x, y, M, K, act);
}

// streaming layer (net2 final, 372MB): non-temporal weight loads
static inline void mv_stream(const float* W, const float* b, const float* x, float* y,
                             int M, int K, int act, hipStream_t stream) {
    matvec_wmma_f32<false, true><<<grid_for(M), 256, 0, stream>>>(
        W, nullptr, b, nullptr, 1.0f, x, y, M, K, act);
}

// net3 layer: W' = W + dW, bias = 2*b + pb
static inline void mv_dw(const float* W, const float* dW, const float* b,
                         const float* pb, const float* x, float* y,
                         int M, int K, int act, hipStream_t stream) {
    matvec_wmma_f32<true, false><<<grid_for(M), 256, 0, stream>>>(
        W, dW, b, pb, 2.0f, x, y, M, K, act);
}

extern "C" void kernel_launch(void* const* d_in, const int* in_sizes, int n_in,
                              void* d_out, int out_size, void* d_ws, size_t ws_size,
                              hipStream_t stream)
{
    (void)in_sizes; (void)n_in; (void)out_size; (void)ws_size;
    const float* inputs = (const float*)d_in[0];
    const float* prev_output = (const float*)d_in[1];
    const float* state = (const float*)d_in[2];
    const float* pwc = (const float*)d_in[3];   // prev_weight_change [W_SZ]
    const float* pbc = (const float*)d_in[4];   // prev_bias_change   [B_SZ]
    const float* w1[3] = {(const float*)d_in[5], (const float*)d_in[7], (const float*)d_in[9]};
    const float* b1[3] = {(const float*)d_in[6], (const float*)d_in[8], (const float*)d_in[10]};
    const float* w2[5] = {(const float*)d_in[11], (const float*)d_in[13], (const float*)d_in[15],
                          (const float*)d_in[17], (const float*)d_in[19]};
    const float* b2[5] = {(const float*)d_in[12], (const float*)d_in[14], (const float*)d_in[16],
                          (const float*)d_in[18], (const float*)d_in[20]};
    const float* w3[5] = {(const float*)d_in[21], (const float*)d_in[23], (const float*)d_in[25],
                          (const float*)d_in[27], (const float*)d_in[29]};
    const float* b3[5] = {(const float*)d_in[22], (const float*)d_in[24], (const float*)d_in[26],
                          (const float*)d_in[28], (const float*)d_in[30]};

    float* out = (float*)d_out;  // [64] out | [128] new_state | weight_change | bias_change
    float* ws  = (float*)d_ws;

    // scratch layout (floats)
    float* X   = ws + 0;     // 256  concat input
    float* h2a = ws + 256;   // 256
    float* h2b = ws + 512;   // 1024
    float* h2c = ws + 1536;  // 1024
    float* h4  = ws + 2560;  // 64
    float* h1a = ws + 2624;  // 1024
    float* h1b = ws + 3648;  // 1024
    float* h3a = ws + 4672;  // 256
    float* h3b = ws + 4928;  // 1024
    float* h3c = ws + 5952;  // 1024
    float* h3d = ws + 6976;  // 64

    // dW / pbc slice offsets for net3 layers
    const size_t wOff[5] = {0, 65536, 327680, 1376256, 1441792};
    const int    bOff[5] = {0, 256, 1280, 2304, 2368};
    const int BIG_M = 1449984 + 2496;  // W_SZ + B_SZ = 1,452,480

    concat_kernel<<<1, 256, 0, stream>>>(inputs, prev_output, state, X);

    // --- net2: small layers then the 372MB streaming layer (tanh*5) ---
    mv(w2[0], b2[0], X,   h2a, 256,  256,  1, stream);
    mv(w2[1], b2[1], h2a, h2b, 1024, 256,  1, stream);
    mv(w2[2], b2[2], h2b, h2c, 1024, 1024, 1, stream);
    mv(w2[3], b2[3], h2c, h4,  64,   1024, 1, stream);
    mv_stream(w2[4], b2[4], h4, out + 192, BIG_M, 64, 3, stream);

    // --- net1 -> out[0:64] ---
    mv(w1[0], b1[0], X,   h1a, 1024, 256,  1, stream);
    mv(w1[1], b1[1], h1a, h1b, 1024, 1024, 1, stream);
    mv(w1[2], b1[2], h1b, out, 64,   1024, 0, stream);

    // --- net3 (W'=w3+dW, bias = 2*b3 + pbc) -> new_state = 2*tanh -> out[64:192] ---
    mv_dw(w3[0], pwc + wOff[0], b3[0], pbc + bOff[0], X,   h3a, 256,  256,  1, stream);
    mv_dw(w3[1], pwc + wOff[1], b3[1], pbc + bOff[1], h3a, h3b, 1024, 256,  1, stream);
    mv_dw(w3[2], pwc + wOff[2], b3[2], pbc + bOff[2], h3b, h3c, 1024, 1024, 1, stream);
    mv_dw(w3[3], pwc + wOff[3], b3[3], pbc + bOff[3], h3c, h3d, 64,   1024, 1, stream);
    mv_dw(w3[4], pwc + wOff[4], b3[4], pbc + bOff[4], h3d, out + 64, 128, 64, 2, stream);
}